// DeepStreamOutput_29119878267615
// MI455X (gfx1250) — compile-verified
//
#include <hip/hip_runtime.h>
#include <math.h>

typedef float v2f __attribute__((ext_vector_type(2)));
typedef float v8f __attribute__((ext_vector_type(8)));

#define B_       2
#define N_       8400
#define NC_      80
#define P_       32
#define HP_      160
#define HW_      (HP_*HP_)       // 25600
#define MAXDET   100
#define ELEMS    (N_*NC_)        // 672000 per batch
#define OUTROW   (4+1+1+HW_)     // 25606
#define NBINS    4096
#define CAP      4096
#define MPAD     112             // 100 rounded up to 16
#define MT       (MPAD/16)       // 7
#define NTILES   (HW_/16)        // 1600

// ---- workspace layout (bytes) ----
#define WS_HIST  0                                    // B*4096 u32
#define WS_CNT   (WS_HIST + B_*NBINS*4)               // B u32
#define WS_THR   (WS_CNT + B_*4)                      // B u32
#define WS_CAND  (WS_THR + B_*4)                      // B*CAP u64 (32784, 16-aligned)
#define WS_A     (WS_CAND + B_*CAP*8)                 // B*112*32 f32
#define WS_BOX   (WS_A + B_*MPAD*P_*4)                // B*100*4 f32
#define WS_M     (WS_BOX + B_*MAXDET*4*4 + 16)        // B*MPAD*25600 f32 (padded rows)

__device__ __forceinline__ unsigned key_of(float f) {
    unsigned u = __float_as_uint(f);
    return (u & 0x80000000u) ? ~u : (u | 0x80000000u);
}
__device__ __forceinline__ float unkey(unsigned k) {
    unsigned u = (k & 0x80000000u) ? (k ^ 0x80000000u) : ~k;
    return __uint_as_float(u);
}

// ---------- 0: zero histogram + counters ----------
__global__ void k_zero(unsigned* p, int n) {
    int i = blockIdx.x * blockDim.x + threadIdx.x;
    if (i < n) p[i] = 0u;
}

// ---------- 1: histogram of ordered keys (top 12 bits) ----------
__global__ void k_hist(const float* __restrict__ scores, unsigned* __restrict__ hist) {
    int gid = blockIdx.x * blockDim.x + threadIdx.x;
    if (gid >= B_ * ELEMS) return;
    int b = gid / ELEMS;
    unsigned key = key_of(scores[gid]);
    atomicAdd(&hist[b * NBINS + (key >> 20)], 1u);
}

// ---------- 2: find threshold key (bin containing 100th largest) ----------
__global__ void k_thresh(const unsigned* __restrict__ hist, unsigned* __restrict__ thr) {
    int b = threadIdx.x;
    if (b >= B_) return;
    unsigned acc = 0, t = 0;
    for (int bin = NBINS - 1; bin >= 0; --bin) {
        acc += hist[b * NBINS + bin];
        if (acc >= MAXDET) { t = ((unsigned)bin) << 20; break; }
    }
    thr[b] = t;
}

// ---------- 3: compact candidates >= threshold ----------
__global__ void k_compact(const float* __restrict__ scores,
                          const unsigned* __restrict__ thr,
                          unsigned* __restrict__ cnt,
                          unsigned long long* __restrict__ cand) {
    int gid = blockIdx.x * blockDim.x + threadIdx.x;
    if (gid >= B_ * ELEMS) return;
    int b = gid / ELEMS;
    unsigned idx = (unsigned)(gid % ELEMS);
    unsigned key = key_of(scores[gid]);
    if (key >= thr[b]) {
        unsigned pos = atomicAdd(&cnt[b], 1u);
        if (pos < CAP)
            cand[b * CAP + pos] = ((unsigned long long)key << 32) | (unsigned)(~idx);
    }
}

// ---------- 4: bitonic top-100 sort + gather boxes/masks ----------
__global__ void __launch_bounds__(1024)
k_sort_gather(const float* __restrict__ boxes, const float* __restrict__ masks,
              const unsigned* __restrict__ cnt, const unsigned long long* __restrict__ cand,
              float* __restrict__ out, float* __restrict__ A_ws, float* __restrict__ box_ws) {
    __shared__ unsigned long long s[CAP];
    int b = blockIdx.x;
    int tid = threadIdx.x;
    unsigned n = cnt[b]; if (n > CAP) n = CAP;
    for (int i = tid; i < CAP; i += blockDim.x)
        s[i] = (i < (int)n) ? cand[b * CAP + i] : 0ull;
    __syncthreads();
    // bitonic sort, descending
    for (int k = 2; k <= CAP; k <<= 1) {
        for (int j = k >> 1; j > 0; j >>= 1) {
            for (int i = tid; i < CAP; i += blockDim.x) {
                int ixj = i ^ j;
                if (ixj > i) {
                    unsigned long long a = s[i], c = s[ixj];
                    bool descBlock = ((i & k) == 0);
                    if (descBlock ? (a < c) : (a > c)) { s[i] = c; s[ixj] = a; }
                }
            }
            __syncthreads();
        }
    }
    if (tid < MAXDET) {
        unsigned long long pk = s[tid];
        unsigned key = (unsigned)(pk >> 32);
        unsigned idx = ~(unsigned)(pk & 0xFFFFFFFFu);
        float logit = unkey(key);
        float val = 1.0f / (1.0f + __expf(-logit));
        int box = (int)(idx / NC_);
        int label = (int)(idx % NC_);
        const float* bx = boxes + ((size_t)b * N_ + box) * 4;
        float cx = bx[0], cy = bx[1], w = bx[2], h = bx[3];
        float x1 = (cx - 0.5f * w) * 640.0f, y1 = (cy - 0.5f * h) * 640.0f;
        float x2 = (cx + 0.5f * w) * 640.0f, y2 = (cy + 0.5f * h) * 640.0f;
        size_t ob = ((size_t)b * MAXDET + tid) * OUTROW;
        out[ob + 0] = x1; out[ob + 1] = y1; out[ob + 2] = x2; out[ob + 3] = y2;
        out[ob + 4] = val; out[ob + 5] = (float)label;
        float* bw = box_ws + (b * MAXDET + tid) * 4;
        bw[0] = x1; bw[1] = y1; bw[2] = x2; bw[3] = y2;
        const float* ms = masks + ((size_t)b * N_ + box) * P_;
        float* aw = A_ws + (b * MPAD + tid) * P_;
        #pragma unroll
        for (int p = 0; p < P_; ++p) aw[p] = ms[p];
    } else if (tid < MPAD) {
        float* aw = A_ws + (b * MPAD + tid) * P_;
        #pragma unroll
        for (int p = 0; p < P_; ++p) aw[p] = 0.0f;
    }
}

// ---------- 5: WMMA GEMM  M[b,t,hw] = sum_p A[b,t,p] * protos[b,p,hw] ----------
// One wave owns one 16-column N-tile; B panel (32x16) kept in registers and
// reused across all 7 M-tiles. C stored unconditionally into padded M_ws.
__global__ void __launch_bounds__(256)
k_gemm(const float* __restrict__ protos, const float* __restrict__ A_ws,
       float* __restrict__ M_ws) {
    int waveG = (int)((blockIdx.x * blockDim.x + threadIdx.x) >> 5);
    int lane = threadIdx.x & 31;
    if (waveG >= B_ * NTILES) return;
    int b  = waveG / NTILES;
    int n0 = (waveG % NTILES) * 16;
    int lhalf = lane >> 4;      // 0: lanes 0-15, 1: lanes 16-31
    int l15   = lane & 15;

    // Load B panel once: per k-step of 4, this lane holds rows (k0+lhalf*2, +1)
    // at column n0+l15.
    const float* Bp = protos + (size_t)b * P_ * HW_ + (n0 + l15);
    v2f bb[8];
    #pragma unroll
    for (int ks = 0; ks < 8; ++ks) {
        bb[ks].x = Bp[(size_t)(ks * 4 + lhalf * 2 + 0) * HW_];
        bb[ks].y = Bp[(size_t)(ks * 4 + lhalf * 2 + 1) * HW_];
    }

    for (int mt = 0; mt < MT; ++mt) {
        int t0 = mt * 16;
        const float* Ap = A_ws + (b * MPAD + t0 + l15) * P_;
        v8f c = {};
        #pragma unroll
        for (int ks = 0; ks < 8; ++ks) {
            v2f a;
            a.x = Ap[ks * 4 + lhalf * 2 + 0];
            a.y = Ap[ks * 4 + lhalf * 2 + 1];
            c = __builtin_amdgcn_wmma_f32_16x16x4_f32(false, a, false, bb[ks],
                                                      (short)0, c, false, false);
        }
        // Store: VGPR r, lane half -> row t0 + lhalf*8 + r, col n0 + l15.
        float* Mp = M_ws + ((size_t)(b * MPAD + t0 + lhalf * 8)) * HW_ + n0 + l15;
        #pragma unroll
        for (int r = 0; r < 8; ++r)
            Mp[(size_t)r * HW_] = c[r];
    }
}

// ---------- 6: bilinear ROI sample + bias ----------
__global__ void k_sample(const float* __restrict__ box_ws, const float* __restrict__ M_ws,
                         const float* __restrict__ bias, float* __restrict__ out) {
    int gid = blockIdx.x * blockDim.x + threadIdx.x;
    if (gid >= B_ * MAXDET * HW_) return;
    int pix = gid % HW_;
    int td  = gid / HW_;            // b*100+det
    int b = td / MAXDET, det = td % MAXDET;
    int h = pix / HP_, w = pix % HP_;
    const float* bx = box_ws + td * 4;
    float x1 = bx[0] * 0.25f, y1 = bx[1] * 0.25f;
    float x2 = bx[2] * 0.25f, y2 = bx[3] * 0.25f;
    float bin_h = (y2 - y1) * (1.0f / HP_);
    float bin_w = (x2 - x1) * (1.0f / HP_);
    float ys = y1 + ((float)h + 0.5f) * bin_h - 0.5f;
    float xs = x1 + ((float)w + 0.5f) * bin_w - 0.5f;
    ys = fminf(fmaxf(ys, 0.0f), (float)(HP_ - 1));
    xs = fminf(fmaxf(xs, 0.0f), (float)(HP_ - 1));
    int y0 = (int)floorf(ys), x0 = (int)floorf(xs);
    int y1i = y0 + 1; if (y1i > HP_ - 1) y1i = HP_ - 1;
    int x1i = x0 + 1; if (x1i > HP_ - 1) x1i = HP_ - 1;
    float ly = ys - (float)y0, lx = xs - (float)x0;
    const float* Mp = M_ws + (size_t)(b * MPAD + det) * HW_;
    float v00 = Mp[y0 * HP_ + x0], v01 = Mp[y0 * HP_ + x1i];
    float v10 = Mp[y1i * HP_ + x0], v11 = Mp[y1i * HP_ + x1i];
    float val = (1.0f - ly) * (1.0f - lx) * v00 + (1.0f - ly) * lx * v01 +
                ly * (1.0f - lx) * v10 + ly * lx * v11 + bias[0];
    out[(size_t)td * OUTROW + 6 + pix] = val;
}

extern "C" void kernel_launch(void* const* d_in, const int* in_sizes, int n_in,
                              void* d_out, int out_size, void* d_ws, size_t ws_size,
                              hipStream_t stream) {
    const float* boxes  = (const float*)d_in[0];
    const float* scores = (const float*)d_in[1];
    const float* protos = (const float*)d_in[2];
    const float* masks  = (const float*)d_in[3];
    const float* bias   = (const float*)d_in[4];
    float* out = (float*)d_out;

    char* ws = (char*)d_ws;
    unsigned*           hist   = (unsigned*)(ws + WS_HIST);
    unsigned*           cnt    = (unsigned*)(ws + WS_CNT);
    unsigned*           thr    = (unsigned*)(ws + WS_THR);
    unsigned long long* cand   = (unsigned long long*)(ws + WS_CAND);
    float*              A_ws   = (float*)(ws + WS_A);
    float*              box_ws = (float*)(ws + WS_BOX);
    float*              M_ws   = (float*)(ws + WS_M);

    // 0: zero hist + counters + thresholds (contiguous u32 region)
    int nz = B_ * NBINS + 2 * B_;
    k_zero<<<(nz + 255) / 256, 256, 0, stream>>>(hist, nz);

    // 1: histogram
    int ne = B_ * ELEMS;
    k_hist<<<(ne + 255) / 256, 256, 0, stream>>>(scores, hist);

    // 2: threshold per batch
    k_thresh<<<1, 32, 0, stream>>>(hist, thr);

    // 3: compact candidates
    k_compact<<<(ne + 255) / 256, 256, 0, stream>>>(scores, thr, cnt, cand);

    // 4: sort + gather (one block per batch)
    k_sort_gather<<<B_, 1024, 0, stream>>>(boxes, masks, cnt, cand, out, A_ws, box_ws);

    // 5: WMMA GEMM: one wave per N-tile, B panel reused across 7 M-tiles
    int nwaves = B_ * NTILES;               // 3200 waves
    k_gemm<<<(nwaves + 7) / 8, 256, 0, stream>>>(protos, A_ws, M_ws);

    // 6: bilinear sample + bias into output
    int np = B_ * MAXDET * HW_;
    k_sample<<<(np + 255) / 256, 256, 0, stream>>>(box_ws, M_ws, bias, out);
}